// GRU_12764642803875
// MI455X (gfx1250) — compile-verified
//
#include <hip/hip_runtime.h>
#include <math.h>

typedef __attribute__((ext_vector_type(2))) float v2f;
typedef __attribute__((ext_vector_type(4))) float v4f;
typedef __attribute__((ext_vector_type(8))) float v8f;

#define B_     8
#define S_     1024
#define IN_    1024
#define STATE_ 1024
#define OUT_   1024
#define H_     8
#define D_     128
#define G3_    (3 * STATE_)

__device__ __forceinline__ float sigmoidf_(float x) {
    return 1.0f / (1.0f + __expf(-x));
}

// CDNA5 async global->LDS copy (VGLOBAL GLOBAL_LOAD_ASYNC_TO_LDS_B128,
// tracked by ASYNCcnt). lds_off = workgroup-relative LDS byte offset
// (low 32 bits of the generic shared-memory address).
__device__ __forceinline__ void async_b128(unsigned lds_off, const void* gptr) {
    asm volatile("global_load_async_to_lds_b128 %0, %1, off"
                 :: "v"(lds_off), "v"(gptr) : "memory");
}
#define WAIT_ASYNC_0() asm volatile("s_wait_asynccnt 0x0" ::: "memory")
#define WAIT_ASYNC_3() asm volatile("s_wait_asynccnt 0x3" ::: "memory")

__device__ __forceinline__ unsigned lds_off_of(const void* p) {
    return (unsigned)(size_t)p;
}

// ---------------------------------------------------------------------------
// GEMM: C[M,N] = A[M,K] @ B[K,N] (+ bias[N]), f32 via V_WMMA_F32_16X16X4_F32
// Block: 256 threads = 8 waves (4 M x 2 N). Block tile 64x128, wave tile 16x64.
// K staged through a double-buffered LDS tile filled with async-to-LDS loads,
// overlapped with WMMA compute on the other buffer.
// ---------------------------------------------------------------------------
#define BM 64
#define BN 128
#define BK 16
#define ASTR 20    // row stride: 16B-aligned quads, 20*l15 mod 64 all distinct
#define BSTR 136   // 2*136 % 64 == 16 -> lane halves hit disjoint banks

__device__ __forceinline__ void gemm_issue(const float* __restrict__ A,
                                           const float* __restrict__ Bm,
                                           int k0, float* AsBuf, float* BsBuf,
                                           int tid, int bm, int bn, int K, int N)
{
    {   // A tile 64x16 : one b128 per thread
        int r = tid >> 2, q = tid & 3;
        const float* g = A + (size_t)(bm + r) * K + k0 + q * 4;
        async_b128(lds_off_of(&AsBuf[r * ASTR + q * 4]), g);
    }
    #pragma unroll
    for (int i = 0; i < 2; ++i) {   // B tile 16x128 : two b128 per thread
        int idx = tid + i * 256;
        int r = idx >> 5, cq = idx & 31;
        const float* g = Bm + (size_t)(k0 + r) * N + bn + cq * 4;
        async_b128(lds_off_of(&BsBuf[r * BSTR + cq * 4]), g);
    }
}

__global__ __launch_bounds__(256)
void gemm_f32_wmma(const float* __restrict__ A, const float* __restrict__ Bm,
                   const float* __restrict__ bias, float* __restrict__ C,
                   int M, int N, int K)
{
    __shared__ float As[2][BM * ASTR];
    __shared__ float Bs[2][BK * BSTR];

    const int tid  = threadIdx.x;
    const int wave = tid >> 5;
    const int lane = tid & 31;
    const int l15  = lane & 15;
    const int half = lane >> 4;
    const int wm   = wave >> 1;      // 0..3
    const int wn   = wave & 1;       // 0..1
    const int bm   = blockIdx.y * BM;
    const int bn   = blockIdx.x * BN;
    const int NC   = K / BK;

    v8f acc[4];
    #pragma unroll
    for (int i = 0; i < 4; ++i)
        #pragma unroll
        for (int j = 0; j < 8; ++j) acc[i][j] = 0.0f;

    gemm_issue(A, Bm, 0, As[0], Bs[0], tid, bm, bn, K, N);

    for (int c = 0; c < NC; ++c) {
        const int cur = c & 1;
        if (c + 1 < NC) {
            gemm_issue(A, Bm, (c + 1) * BK, As[cur ^ 1], Bs[cur ^ 1],
                       tid, bm, bn, K, N);
            WAIT_ASYNC_3();          // 3 older ops (current chunk) complete
        } else {
            WAIT_ASYNC_0();
        }
        __syncthreads();

        const float* Asb = As[cur];
        const float* Bsb = Bs[cur];
        const int arow = wm * 16 + l15;
        #pragma unroll
        for (int kk = 0; kk < 4; ++kk) {
            const int kb = kk * 4 + half * 2;      // K pair per lane half
            v2f a;
            a[0] = Asb[arow * ASTR + kb];
            a[1] = Asb[arow * ASTR + kb + 1];
            #pragma unroll
            for (int nt = 0; nt < 4; ++nt) {
                const int nc = wn * 64 + nt * 16 + l15;
                v2f b;
                b[0] = Bsb[kb * BSTR + nc];
                b[1] = Bsb[(kb + 1) * BSTR + nc];
                acc[nt] = __builtin_amdgcn_wmma_f32_16x16x4_f32(
                    false, a, false, b, (short)0, acc[nt], false, false);
            }
        }
        __syncthreads();             // reads done before buffer is re-filled
    }

    // write back: C/D layout: VGPR j -> M = j (lanes 0-15) / j+8 (16-31)
    #pragma unroll
    for (int nt = 0; nt < 4; ++nt) {
        const int col = bn + wn * 64 + nt * 16 + l15;
        const float bv = bias ? bias[col] : 0.0f;
        const int r0 = bm + wm * 16 + half * 8;
        #pragma unroll
        for (int j = 0; j < 8; ++j)
            C[(size_t)(r0 + j) * N + col] = acc[nt][j] + bv;
    }
}

// ---------------------------------------------------------------------------
// Recurrent scan. One workgroup per head; all 3 weight matrices (~204 KB with
// bank padding) live in LDS for the whole scan (CDNA5: 320 KB/WGP). The 8
// batch rows are batched into the WMMA M dimension (rows 8..15 zero pad).
// 8 waves, 16 output columns each. Phase-2 weight fragments are loop-invariant
// per wave and are hoisted into 64 VGPRs before the time loop. Per step:
//   init accs from LDS-prefetched inputs -> issue async prefetch for t+1
//   -> phase1 (R,F share A fragments) -> sync -> phase2 (register-resident Wc)
//   -> s_wait_asynccnt 0 -> sync.
// ---------------------------------------------------------------------------
#define WSTR 136   // weight LDS row stride: lane halves on disjoint banks
#define HSTR 132   // h/rh LDS row stride: conflict-free A-fragment reads

__device__ __forceinline__ void scan_issue_inputs(const float* __restrict__ proj,
                                                  int t, int head,
                                                  float* dst /*3072 floats*/,
                                                  int tid)
{
    #pragma unroll
    for (int i = 0; i < 3; ++i) {
        int q = tid + i * 256;       // quad index 0..767
        int e = q * 4;               // element 0..3071 (gate-major)
        int g = e >> 10;             // gate: 0=z, 1=f, 2=r
        int rem = e & 1023;
        int b = rem >> 7;
        int c = rem & 127;
        const float* gp = proj + ((size_t)b * S_ + t) * G3_
                               + g * STATE_ + head * D_ + c;
        async_b128(lds_off_of(dst + e), gp);
    }
}

__global__ __launch_bounds__(256)
void gru_scan(const float* __restrict__ proj,   // [B,S,3*STATE]
              const float* __restrict__ h0,     // [B,STATE]
              const float* __restrict__ Wc,     // state_weight  [H,D,D]
              const float* __restrict__ Wf,     // forget_weight [H,D,D]
              const float* __restrict__ Wr,     // reset_weight  [H,D,D]
              float* __restrict__ hs,           // [B,S,STATE]
              float* __restrict__ hT)           // [B,STATE]
{
    __shared__ float Wr_s[D_ * WSTR];
    __shared__ float Wf_s[D_ * WSTR];
    __shared__ float Wc_s[D_ * WSTR];
    __shared__ float h_s [16 * HSTR];
    __shared__ float rh_s[16 * HSTR];
    __shared__ float f_s [B_ * D_];
    __shared__ float in_s[2][G3_];    // double-buffered z/f/r inputs

    const int head = blockIdx.x;
    const int tid  = threadIdx.x;
    const int wave = tid >> 5;
    const int lane = tid & 31;
    const int l15  = lane & 15;
    const int half = lane >> 4;
    const int n0   = wave * 16;       // 16 output columns per wave

    // prefetch inputs for t=0 (overlaps the weight staging below)
    scan_issue_inputs(proj, 0, head, in_s[0], tid);

    // stage this head's 3 weight matrices into LDS via async copies
    const size_t wb = (size_t)head * D_ * D_;
    for (int i = tid; i < (D_ * D_) / 4; i += 256) {   // 16 iters x 3 issues
        int r = i >> 5, q = i & 31;
        const size_t gofs = wb + (size_t)r * D_ + q * 4;
        async_b128(lds_off_of(&Wr_s[r * WSTR + q * 4]), Wr + gofs);
        async_b128(lds_off_of(&Wf_s[r * WSTR + q * 4]), Wf + gofs);
        async_b128(lds_off_of(&Wc_s[r * WSTR + q * 4]), Wc + gofs);
    }
    // init h (rows 8..15 = 0) and rh pad rows
    for (int i = tid; i < 16 * D_; i += 256) {
        int r = i >> 7, c = i & (D_ - 1);
        float v = (r < B_) ? h0[(size_t)r * STATE_ + head * D_ + c] : 0.0f;
        h_s [r * HSTR + c] = v;
        rh_s[r * HSTR + c] = 0.0f;
    }
    WAIT_ASYNC_0();
    __syncthreads();

    // hoist phase-2 (Wc) B-fragments: loop-invariant across all 1024 steps
    v2f wc[32];
    #pragma unroll
    for (int kk = 0; kk < 32; ++kk) {
        const int kb = kk * 4 + half * 2;
        wc[kk][0] = Wc_s[kb * WSTR + n0 + l15];
        wc[kk][1] = Wc_s[(kb + 1) * WSTR + n0 + l15];
    }

    for (int t = 0; t < S_; ++t) {
        const float* in_cur = in_s[t & 1];

        // ---- init accumulators with projected inputs (C fragment layout) ----
        v8f accR, accF, accC;
        #pragma unroll
        for (int j = 0; j < 8; ++j) {
            if (half == 0) {   // rows 0..7 = real batches
                const int e = j * D_ + n0 + l15;
                accC[j] = in_cur[e];                  // z
                accF[j] = in_cur[STATE_ + e];         // f
                accR[j] = in_cur[2 * STATE_ + e];     // r
            } else {           // rows 8..15 = zero pad
                accC[j] = 0.0f; accF[j] = 0.0f; accR[j] = 0.0f;
            }
        }

        // prefetch next step's inputs; lands while we do both matmul phases
        if (t + 1 < S_)
            scan_issue_inputs(proj, t + 1, head, in_s[(t + 1) & 1], tid);

        // ---- phase 1: R += h@Wr, F += h@Wf (shared A fragment) ----
        #pragma unroll
        for (int k = 0; k < D_; k += 4) {
            const int kb = k + half * 2;
            v2f a;
            a[0] = h_s[l15 * HSTR + kb];
            a[1] = h_s[l15 * HSTR + kb + 1];
            v2f br, bf;
            br[0] = Wr_s[kb * WSTR + n0 + l15];
            br[1] = Wr_s[(kb + 1) * WSTR + n0 + l15];
            bf[0] = Wf_s[kb * WSTR + n0 + l15];
            bf[1] = Wf_s[(kb + 1) * WSTR + n0 + l15];
            accR = __builtin_amdgcn_wmma_f32_16x16x4_f32(
                false, a, false, br, (short)0, accR, false, false);
            accF = __builtin_amdgcn_wmma_f32_16x16x4_f32(
                false, a, false, bf, (short)0, accF, false, false);
        }
        if (half == 0) {
            #pragma unroll
            for (int j = 0; j < 8; ++j) {
                const int c = n0 + l15;
                const float r = sigmoidf_(accR[j]);
                const float f = sigmoidf_(accF[j]);
                rh_s[j * HSTR + c] = r * h_s[j * HSTR + c];
                f_s [j * D_   + c] = f;
            }
        }
        __syncthreads();

        // ---- phase 2: C += (r*h)@Wc with register-resident weights ----
        #pragma unroll
        for (int kk = 0; kk < 32; ++kk) {
            const int kb = kk * 4 + half * 2;
            v2f a;
            a[0] = rh_s[l15 * HSTR + kb];
            a[1] = rh_s[l15 * HSTR + kb + 1];
            accC = __builtin_amdgcn_wmma_f32_16x16x4_f32(
                false, a, false, wc[kk], (short)0, accC, false, false);
        }
        if (half == 0) {
            #pragma unroll
            for (int j = 0; j < 8; ++j) {
                const int c = n0 + l15;
                const float cc = tanhf(accC[j]);
                const float f  = f_s[j * D_ + c];
                const float hn = f * h_s[j * HSTR + c] + (1.0f - f) * cc;
                h_s[j * HSTR + c] = hn;
                hs[((size_t)j * S_ + t) * STATE_ + head * D_ + c] = hn;
            }
        }
        WAIT_ASYNC_0();          // next step's inputs have landed
        __syncthreads();         // h_s + in_s visible to all waves
    }

    // ---- final state ----
    if (half == 0) {
        #pragma unroll
        for (int j = 0; j < 8; ++j)
            hT[(size_t)j * STATE_ + head * D_ + n0 + l15] =
                h_s[j * HSTR + n0 + l15];
    }
}

// ---------------------------------------------------------------------------
extern "C" void kernel_launch(void* const* d_in, const int* in_sizes, int n_in,
                              void* d_out, int out_size, void* d_ws, size_t ws_size,
                              hipStream_t stream)
{
    const float* x  = (const float*)d_in[0];   // [B,S,IN]
    const float* h0 = (const float*)d_in[1];   // [B,STATE]
    const float* Wi = (const float*)d_in[2];   // [IN,3*STATE]
    const float* bi = (const float*)d_in[3];   // [3*STATE]
    const float* Wc = (const float*)d_in[4];   // state_weight  [H,D,D]
    const float* Wf = (const float*)d_in[5];   // forget_weight [H,D,D]
    const float* Wr = (const float*)d_in[6];   // reset_weight  [H,D,D]
    const float* Wo = (const float*)d_in[7];   // [STATE,OUT]

    float* out = (float*)d_out;                         // [B,S,OUT]
    float* hT  = out + (size_t)B_ * S_ * OUT_;          // [B,STATE]

    float* proj = (float*)d_ws;                         // [B,S,3*STATE]
    float* hs   = proj + (size_t)B_ * S_ * G3_;         // [B,S,STATE]

    dim3 blk(256);

    // 1) proj = x @ Wi + bi
    gemm_f32_wmma<<<dim3(G3_ / BN, (B_ * S_) / BM), blk, 0, stream>>>(
        x, Wi, bi, proj, B_ * S_, G3_, IN_);

    // 2) recurrent scan (weights resident in LDS, inputs async-prefetched)
    gru_scan<<<dim3(H_), blk, 0, stream>>>(proj, h0, Wc, Wf, Wr, hs, hT);

    // 3) out = hs @ Wo
    gemm_f32_wmma<<<dim3(OUT_ / BN, (B_ * S_) / BM), blk, 0, stream>>>(
        hs, Wo, nullptr, out, B_ * S_, OUT_, STATE_);
}